// SageRouter_89979564851201
// MI455X (gfx1250) — compile-verified
//
#include <hip/hip_runtime.h>
#include <hip/hip_bf16.h>
#include <math.h>

typedef __attribute__((ext_vector_type(2))) float v2f;
typedef __attribute__((ext_vector_type(4))) float v4f;
typedef __attribute__((ext_vector_type(8))) float v8f;

// Problem constants
#define BB   64
#define NN   4096
#define DD   1024
#define HH   256
#define EE   64
#define NCHUNK 64                 // N split across blocks
#define ROWS_PER_BLK (NN / NCHUNK) // 64

// ---------------------------------------------------------------------------
// Kernel 0: zero the agg accumulator (ws is poisoned once, never re-poisoned;
// we must re-zero every call for graph-replay determinism).
// ---------------------------------------------------------------------------
__global__ void k_zero(float* __restrict__ p, int n4) {
  int i = blockIdx.x * blockDim.x + threadIdx.x;
  if (i < n4) {
    v4f z = {0.f, 0.f, 0.f, 0.f};
    ((v4f*)p)[i] = z;
  }
}

// ---------------------------------------------------------------------------
// Kernel 1: streaming mean over N.  This is the only HBM-bound stage (1 GiB
// of x).  b128 non-temporal loads (stream-once data, keep L2 for agg/q_w),
// register accumulation, then scaled f32 atomics into agg.
// ---------------------------------------------------------------------------
__global__ __launch_bounds__(256) void k_mean(const float* __restrict__ x,
                                              float* __restrict__ agg) {
  const int blk   = blockIdx.x;
  const int b     = blk >> 6;       // / NCHUNK
  const int chunk = blk & (NCHUNK - 1);
  const int tid   = threadIdx.x;    // 256 threads * float4 = 1024 floats = D

  const size_t base = ((size_t)b * NN + (size_t)chunk * ROWS_PER_BLK) * DD
                      + (size_t)tid * 4;
  const v4f* p = (const v4f*)(x + base);

  v4f acc = {0.f, 0.f, 0.f, 0.f};
#pragma unroll 8
  for (int n = 0; n < ROWS_PER_BLK; ++n) {
    v4f v = __builtin_nontemporal_load(p + (size_t)n * (DD / 4));
    acc += v;
  }

  const float inv = 1.0f / (float)NN;
  float* dst = agg + b * DD + tid * 4;
  atomicAdd(dst + 0, acc.x * inv);
  atomicAdd(dst + 1, acc.y * inv);
  atomicAdd(dst + 2, acc.z * inv);
  atomicAdd(dst + 3, acc.w * inv);
}

// ---------------------------------------------------------------------------
// Kernel 2: query = agg @ q_w^T + q_b   (64x1024 @ 1024x256 -> 64x256)
// Pure-fp32 WMMA: V_WMMA_F32_16X16X4_F32, one wave per 16x16 tile.
//   A lane layout: M = lane%16, K = 2*(lane/16)+{0,1}  (8B contiguous load)
//   B lane layout: N = lane%16, K = 2*(lane/16)+{0,1}; B[k][n] = q_w[n][k]
//   C/D: VGPR r -> row r + 8*(lane/16), col = lane%16
// ---------------------------------------------------------------------------
__global__ __launch_bounds__(32) void k_query(const float* __restrict__ agg,
                                              const float* __restrict__ q_w,
                                              const float* __restrict__ q_b,
                                              float* __restrict__ query) {
  const int lane  = threadIdx.x;     // 0..31, full wave (EXEC all ones)
  const int lo    = lane & 15;
  const int hi    = lane >> 4;
  const int mtile = blockIdx.x >> 4; // 0..3   (64 rows / 16)
  const int ntile = blockIdx.x & 15; // 0..15  (256 cols / 16)

  const float* arow = agg + (mtile * 16 + lo) * DD + 2 * hi;
  const float* brow = q_w + (ntile * 16 + lo) * DD + 2 * hi;

  v8f c = {};
#pragma unroll 4
  for (int k0 = 0; k0 < DD; k0 += 4) {
    v2f a = *(const v2f*)(arow + k0);
    v2f b = *(const v2f*)(brow + k0);
    c = __builtin_amdgcn_wmma_f32_16x16x4_f32(
        /*neg_a=*/false, a, /*neg_b=*/false, b,
        /*c_mod=*/(short)0, c, /*reuse_a=*/false, /*reuse_b=*/false);
  }

  const int   col = ntile * 16 + lo;
  const float qb  = q_b[col];
#pragma unroll
  for (int r = 0; r < 8; ++r) {
    const int row = mtile * 16 + r + 8 * hi;
    query[row * HH + col] = c[r] + qb;
  }
}

// ---------------------------------------------------------------------------
// Kernel 3: gate, logits, modulation, top-2, softmax, load-balance loss.
// All operands are tiny and L2-hot after kernels 1-2.
// ---------------------------------------------------------------------------
__global__ __launch_bounds__(256) void k_router(const float* __restrict__ agg,
                                                const float* __restrict__ gate_w,
                                                const float* __restrict__ gate_b,
                                                const float* __restrict__ query,
                                                const float* __restrict__ ek,
                                                const float* __restrict__ smask,
                                                float* __restrict__ out) {
  __shared__ float modul[BB * EE];   // 16 KB, later reused as softmax probs
  __shared__ float gs[BB];
  __shared__ float red[EE];

  const int tid = threadIdx.x;

  // shared-expert gate g_s[b] = sigmoid(agg[b] . gate_w + gate_b)
  if (tid < BB) {
    const float* a = agg + tid * DD;
    float s = 0.f;
    for (int i = 0; i < DD; ++i) s = fmaf(a[i], gate_w[i], s);
    s += gate_b[0];
    gs[tid] = 1.f / (1.f + expf(-s));
  }
  __syncthreads();

  // modulated logits: (query @ ek^T)/sqrt(H) + mask*log(g)+(1-mask)*log(1-g)
  const float invT = 1.0f / 16.0f;   // TEMP = sqrt(256)
#pragma unroll
  for (int i = 0; i < (BB * EE) / 256; ++i) {
    const int p = tid + 256 * i;     // 0..4095
    const int b = p >> 6, e = p & 63;
    const float* q = query + b * HH;
    const float* k = ek + e * HH;
    float s = 0.f;
    for (int j = 0; j < HH; ++j) s = fmaf(q[j], k[j], s);
    const float g = gs[b];
    const float m = smask[e];
    modul[p] = s * invT + m * logf(g + 1e-9f) + (1.f - m) * logf(1.f - g + 1e-9f);
  }
  __syncthreads();

  // per-row top-2 (+ 2-way softmax) and in-place full softmax
  if (tid < BB) {
    float* row = modul + tid * EE;
    float m1 = -__builtin_inff(), m2 = -__builtin_inff();
    int   i1 = 0, i2 = 0;
    for (int e = 0; e < EE; ++e) {
      const float v = row[e];
      if (v > m1)      { m2 = m1; i2 = i1; m1 = v; i1 = e; }
      else if (v > m2) { m2 = v;  i2 = e; }
    }
    const float r2  = expf(m2 - m1);
    const float inv = 1.f / (1.f + r2);
    out[tid * 2 + 0]       = (float)i1;   // indices (as output dtype)
    out[tid * 2 + 1]       = (float)i2;
    out[128 + tid * 2 + 0] = inv;         // gating weights
    out[128 + tid * 2 + 1] = r2 * inv;

    float sum = 0.f;
    for (int e = 0; e < EE; ++e) { float t = expf(row[e] - m1); row[e] = t; sum += t; }
    const float rs = 1.f / sum;
    for (int e = 0; e < EE; ++e) row[e] *= rs;
  }
  __syncthreads();

  // P[e] = mean_b probs[b][e];  loss = 0.01 * E * sum(P^2)
  if (tid < EE) {
    float s = 0.f;
    for (int b = 0; b < BB; ++b) s += modul[b * EE + tid];
    const float P = s * (1.0f / (float)BB);
    red[tid] = P * P;
  }
  __syncthreads();
  if (tid == 0) {
    float s = 0.f;
    for (int e = 0; e < EE; ++e) s += red[e];
    out[256] = 0.01f * (float)EE * s;
  }
}

// ---------------------------------------------------------------------------
extern "C" void kernel_launch(void* const* d_in, const int* in_sizes, int n_in,
                              void* d_out, int out_size, void* d_ws, size_t ws_size,
                              hipStream_t stream) {
  const float* x      = (const float*)d_in[0];
  const float* gate_w = (const float*)d_in[1];
  const float* gate_b = (const float*)d_in[2];
  const float* q_w    = (const float*)d_in[3];
  const float* q_b    = (const float*)d_in[4];
  const float* ek     = (const float*)d_in[5];
  const float* smask  = (const float*)d_in[6];
  float* out = (float*)d_out;

  float* agg   = (float*)d_ws;          // 64*1024 floats (256 KB)
  float* query = agg + BB * DD;         // 64*256  floats (64 KB)

  k_zero  <<<64,           256, 0, stream>>>(agg, (BB * DD) / 4);
  k_mean  <<<BB * NCHUNK,  256, 0, stream>>>(x, agg);
  k_query <<<(BB/16)*(HH/16), 32, 0, stream>>>(agg, q_w, q_b, query);
  k_router<<<1,            256, 0, stream>>>(agg, gate_w, gate_b, query, ek, smask, out);
}